// PatchMaker_13649406067421
// MI455X (gfx1250) — compile-verified
//
#include <hip/hip_runtime.h>
#include <cstdint>

// ---------------- problem constants (from reference) ----------------
#define PATCH      16
#define NCH        3
#define HH         512
#define WW         512
#define HP_        32                       // HH / PATCH
#define WP_        32                       // WW / PATCH
#define OUT_TILE   (WP_ * PATCH * PATCH * NCH)  // 24576 floats out per (b,hp) tile
#define NTHREADS   256

// ---------------- padded LDS layout ----------------
// row stride 516 floats: 16B-aligned rows (516*4 = 2064), 516 % 64 = 4 banks/row.
// channel stride 16*516 = 8256. Stride-4 output scan touches 12 distinct ph per
// pw-block; banks = 4*ph + col (mod 64) land on disjoint mod-4 residue classes
// across adjacent pw blocks -> conflict-free gather.
#define RS   516
#define CSZ  (PATCH * RS)        // 8256
#define LDS_FLOATS (NCH * CSZ)   // 24768 floats = 99,072 B (3 blocks / WGP)

typedef int   v4i __attribute__((ext_vector_type(4)));
typedef float v4f __attribute__((ext_vector_type(4)));

typedef __attribute__((address_space(1))) v4i  g_v4i;
typedef __attribute__((address_space(3))) v4i  l_v4i;

// ---------------- CDNA5 async global->LDS path ----------------
#if defined(__has_builtin)
#  if __has_builtin(__builtin_amdgcn_global_load_async_to_lds_b128)
#    define HAVE_ASYNC_B128 1
#  endif
#  if __has_builtin(__builtin_amdgcn_s_wait_asynccnt)
#    define HAVE_WAIT_ASYNC 1
#  endif
#endif

__device__ __forceinline__ void async_copy_b128(const float* gsrc, float* ldst) {
#if defined(HAVE_ASYNC_B128)
  __builtin_amdgcn_global_load_async_to_lds_b128(
      (g_v4i*)gsrc, (l_v4i*)ldst, /*offset=*/0, /*cpol=*/0);
#else
  unsigned lds_off = (unsigned)(uintptr_t)ldst;  // low 32 bits = LDS offset
  asm volatile("global_load_async_to_lds_b128 %0, %1, off"
               :: "v"(lds_off), "v"(gsrc) : "memory");
#endif
}

__device__ __forceinline__ void wait_async_zero() {
#if defined(HAVE_WAIT_ASYNC)
  __builtin_amdgcn_s_wait_asynccnt(0);
#else
  asm volatile("s_wait_asynccnt 0x0" ::: "memory");
#endif
}

// ---------------- kernel ----------------
__global__ __launch_bounds__(NTHREADS)
void patchify_kernel(const float* __restrict__ x, float* __restrict__ out) {
  __shared__ float smem[LDS_FLOATS];

  const int t  = threadIdx.x;
  const int b  = blockIdx.x >> 5;    // / HP_
  const int hp = blockIdx.x & 31;    // % HP_

  // ---- phase 1: global -> LDS via async b128 DMA, fully coalesced ----
  // Each thread owns a fixed (row-parity, 16B-column) slot; every async op's
  // addresses are per-thread base + compile-time constant (minimal VALU).
  const float* src_base = x + (size_t)b * (NCH * HH * WW) + (size_t)hp * (PATCH * WW);
  const int lane_row = t >> 7;              // 0..1   (128 lanes cover one 512-float row)
  const int w4       = (t & 127) << 2;      // column, multiple of 4 floats (16 B)

  const float* gbase = src_base + lane_row * WW + w4;
  float*       lbase = &smem[lane_row * RS + w4];

  #pragma unroll
  for (int c = 0; c < NCH; ++c) {
    #pragma unroll
    for (int rr = 0; rr < PATCH; rr += 2) {       // 8 row-pairs -> 24 async ops total
      async_copy_b128(gbase + c * (HH * WW) + rr * WW,
                      lbase + c * CSZ       + rr * RS);
    }
  }

  wait_async_zero();   // each wave drains its own ASYNCcnt
  __syncthreads();     // then all waves' LDS writes are published

  // ---- phase 2: conflict-free LDS gather -> contiguous float4 NT stores ----
  float* dst = out + (size_t)blockIdx.x * OUT_TILE;

  #pragma unroll 4
  for (int it = 0; it < OUT_TILE / 4 / NTHREADS; ++it) {      // 24 iterations
    const int o = (it * NTHREADS + t) << 2;
    v4f val;
    #pragma unroll
    for (int j = 0; j < 4; ++j) {
      const int oo  = o + j;
      const int col = oo / 48;             // wp*16 + pw (continuous column index)
      const int r2  = oo - col * 48;
      const int ph  = r2 / 3;
      const int c   = r2 - ph * 3;
      val[j] = smem[c * CSZ + ph * RS + col];
    }
    __builtin_nontemporal_store(val, (v4f*)(dst + o));
  }
}

// ---------------- launcher ----------------
extern "C" void kernel_launch(void* const* d_in, const int* in_sizes, int n_in,
                              void* d_out, int out_size, void* d_ws, size_t ws_size,
                              hipStream_t stream) {
  const float* x = (const float*)d_in[0];
  float* out = (float*)d_out;
  const int B = in_sizes[0] / (NCH * HH * WW);   // 64
  dim3 grid(B * HP_);                            // 2048 blocks, one (b,hp) tile each
  patchify_kernel<<<grid, NTHREADS, 0, stream>>>(x, out);
  (void)n_in; (void)out_size; (void)d_ws; (void)ws_size;
}